// Architecture_54451595379019
// MI455X (gfx1250) — compile-verified
//
#include <hip/hip_runtime.h>
#include <hip/hip_bf16.h>
#include <math.h>

typedef __attribute__((ext_vector_type(2))) float v2f;
typedef __attribute__((ext_vector_type(8))) float v8f;

#define HD    16
#define CIN   8
#define TD_C  24
#define HW    256        // 16x16 spatial
#define BATCH 32
#define KDIM  4096       // HID_FLAT
#define N_BU  2048       // IN_FLAT
#define N_TD  6144       // TD_FLAT
#define KC_BU 8          // split-K chunks for bottom-up GEMMs (klen=512)
#define KC_TD 4          // split-K chunks for top-down GEMMs (klen=1024)

__device__ __forceinline__ float sigmoidf_(float x) {
  return 1.0f / (1.0f + __expf(-x));
}
__device__ __forceinline__ v2f relu2(v2f v) {
  v.x = v.x > 0.f ? v.x : 0.f;
  v.y = v.y > 0.f ? v.y : 0.f;
  return v;
}
// non-temporal: weight stream is single-use, keep it from evicting A rows in L2
__device__ __forceinline__ v2f ntload2(const float* p) {
  return __builtin_nontemporal_load((const v2f*)p);
}

// ---------------------------------------------------------------------------
// WMMA fp32 GEMM:  C(M=32, Nout) = scale * f(A)(32,K) @ W(Nout,K)^T [+ bias]
// One wave = one 16x16 tile; 8 waves/block cover a 32x64 panel.
// v_wmma_f32_16x16x4_f32: A 16x4, B 4x16.
//   A layout: lane<16 -> (K+0,K+1), lane>=16 -> (K+2,K+3), row M = lane&15
//   B layout (mirror): lane<16 -> rows K+0,K+1; lane>=16 -> K+2,K+3; col N = lane&15
// Both are contiguous float2 loads per lane.
// K-loop runs in tiles of 32: 16 independent loads issued as a burst (s_clause),
// then 8 WMMAs drain them with incremental s_wait_loadcnt -> deep MLP per wave.
// SPLITK: blockIdx.y = K-chunk; partial sums stored to C[kc * BATCH*ldc + ...],
//         bias added later by reduce_bias_kernel (deterministic, no atomics).
// ---------------------------------------------------------------------------
template<bool RELU_IN, bool RELU_OUT, bool SPLITK, bool SCALED>
__global__ __launch_bounds__(256) void gemm_wmma_f32(
    const float* __restrict__ A, const float* __restrict__ W,
    const float* __restrict__ bias, float* __restrict__ C,
    int Nout, int K, int ldc,
    const float* __restrict__ conn, int scale_idx)
{
  const int w    = threadIdx.x >> 5;
  const int lane = threadIdx.x & 31;
  const int r16  = lane & 15;
  const int hi   = lane >> 4;
  const int m0   = (w & 1) << 4;                      // 0 or 16
  const int n0   = blockIdx.x * 64 + ((w >> 1) << 4); // 4 N-tiles per block

  const int knum = SPLITK ? (int)gridDim.y : 1;
  const int kc   = SPLITK ? (int)blockIdx.y : 0;
  const int klen = K / knum;                           // multiple of 32
  const int kbeg = kc * klen;
  const int kend = kbeg + klen;

  float scale = 1.0f;
  if (SCALED) scale = conn[scale_idx];

  const int arow   = m0 + r16;                   // batch row 0..31
  const int nrow   = n0 + r16;                   // weight row (output feature)
  const int nclamp = nrow < Nout ? nrow : Nout - 1;

  const float* Ap = A + (size_t)arow * K + hi * 2;
  const float* Wp = W + (size_t)nclamp * K + hi * 2;

  v8f acc = {};
  for (int k = kbeg; k < kend; k += 32) {
    v2f av[8], bv[8];
    #pragma unroll
    for (int j = 0; j < 8; ++j)
      av[j] = *(const v2f*)(Ap + k + 4 * j);
    #pragma unroll
    for (int j = 0; j < 8; ++j)
      bv[j] = ntload2(Wp + k + 4 * j);
    #pragma unroll
    for (int j = 0; j < 8; ++j) {
      v2f a = av[j];
      if (RELU_IN) a = relu2(a);
      if (SCALED)  a = a * scale;
      acc = __builtin_amdgcn_wmma_f32_16x16x4_f32(false, a, false, bv[j],
                                                  (short)0, acc, false, false);
    }
  }

  // C/D layout: col = lane&15; lane<16 -> rows m0+0..7, lane>=16 -> rows m0+8..15
  const int ncol = n0 + r16;
  if (ncol < Nout) {
    if (SPLITK) {
      float* Cp = C + (size_t)kc * BATCH * ldc;
      #pragma unroll
      for (int r = 0; r < 8; ++r) {
        const int m = m0 + hi * 8 + r;
        Cp[(size_t)m * ldc + ncol] = acc[r];
      }
    } else {
      const float bn = bias[ncol];
      #pragma unroll
      for (int r = 0; r < 8; ++r) {
        const int m = m0 + hi * 8 + r;
        float v = acc[r] + bn;
        if (RELU_OUT) v = v > 0.f ? v : 0.f;
        C[(size_t)m * ldc + ncol] = v;
      }
    }
  }
}

// C[m,n] = bias[n] + sum_kc part[kc][m,n]   (deterministic split-K reduction)
__global__ void reduce_bias_kernel(const float* __restrict__ part,
                                   const float* __restrict__ bias,
                                   float* __restrict__ C,
                                   int Nout, int ldc, int knum)
{
  const int i = blockIdx.x * blockDim.x + threadIdx.x;
  if (i >= BATCH * Nout) return;
  const int m = i / Nout, n = i % Nout;
  float acc = bias[n];
  for (int kc = 0; kc < knum; ++kc)
    acc += part[(size_t)kc * BATCH * ldc + (size_t)m * ldc + n];
  C[(size_t)m * ldc + n] = acc;
}

// ---------------------------------------------------------------------------
// conv_in: y = conv2d(x[:,0], w, b), in=8ch out=16ch 3x3 SAME, 16x16 spatial.
// ---------------------------------------------------------------------------
__global__ __launch_bounds__(256) void conv_in_kernel(
    const float* __restrict__ x, const float* __restrict__ w,
    const float* __restrict__ bias, float* __restrict__ out)
{
  __shared__ float s[CIN * HW];
  const int b = blockIdx.x;
  const int tid = threadIdx.x;
  const int py = tid >> 4, px = tid & 15;

  #pragma unroll
  for (int c = 0; c < CIN; ++c)
    s[c * HW + tid] = x[((size_t)b * 2 * CIN + c) * HW + tid];  // t=0 slice
  __syncthreads();

  for (int oc = 0; oc < HD; ++oc) {
    float acc = bias[oc];
    const float* wb = w + oc * (CIN * 9);
    for (int ic = 0; ic < CIN; ++ic) {
      const float* wr = wb + ic * 9;
      const float* sp = s + ic * HW;
      #pragma unroll
      for (int ky = 0; ky < 3; ++ky) {
        const int yy = py + ky - 1;
        if (yy < 0 || yy > 15) continue;
        #pragma unroll
        for (int kx = 0; kx < 3; ++kx) {
          const int xx = px + kx - 1;
          if (xx < 0 || xx > 15) continue;
          acc += sp[yy * 16 + xx] * wr[ky * 3 + kx];
        }
      }
    }
    out[(size_t)b * (HD * HW) + oc * HW + tid] = acc;
  }
}

// ---------------------------------------------------------------------------
// Fused ConvGRUTopDown cell. One block per image, one thread per pixel.
//   mod = sigmoid(td); gates = conv(cat(bu,h)*mod); r,u = sigmoid(split)
//   cand = tanh(conv(cat(bu, r*h)*mod)); h' = (1-u)h + u*cand
// Null bu_full -> bu_bias broadcast (structural-zero GEMM input) or zero.
// Null h_in -> zeros. Null td_full -> td_bias broadcast or zeros (last node).
// ---------------------------------------------------------------------------
__global__ __launch_bounds__(256) void cell_kernel(
    const float* __restrict__ bu_full, const float* __restrict__ bu_bias,
    const float* __restrict__ h_in,
    const float* __restrict__ td_full, const float* __restrict__ td_bias,
    const float* __restrict__ gate_w, const float* __restrict__ gate_b,
    const float* __restrict__ cand_w, const float* __restrict__ cand_b,
    float* __restrict__ h_out)
{
  __shared__ float s_stack[TD_C * HW];      // 24 KB
  __shared__ float s_ru[2 * HD * HW];       // 32 KB: r then u
  const int b = blockIdx.x;
  const int tid = threadIdx.x;
  const int py = tid >> 4, px = tid & 15;

  float bu[CIN], h[HD], mod[TD_C];
  #pragma unroll
  for (int c = 0; c < CIN; ++c)
    bu[c] = bu_full ? bu_full[(size_t)b * (CIN * HW) + c * HW + tid]
                    : (bu_bias ? bu_bias[c * HW + tid] : 0.f);
  #pragma unroll
  for (int c = 0; c < HD; ++c)
    h[c] = h_in ? h_in[(size_t)b * (HD * HW) + c * HW + tid] : 0.f;
  #pragma unroll
  for (int c = 0; c < TD_C; ++c) {
    const float t = td_full ? td_full[(size_t)b * (TD_C * HW) + c * HW + tid]
                            : (td_bias ? td_bias[c * HW + tid] : 0.f);
    mod[c] = sigmoidf_(t);
  }

  #pragma unroll
  for (int c = 0; c < TD_C; ++c)
    s_stack[c * HW + tid] = ((c < CIN) ? bu[c] : h[c - CIN]) * mod[c];
  __syncthreads();

  // gate conv: 32 output channels -> r (0..15), u (16..31)
  for (int oc = 0; oc < 2 * HD; ++oc) {
    float acc = gate_b[oc];
    const float* wb = gate_w + oc * (TD_C * 9);
    for (int ic = 0; ic < TD_C; ++ic) {
      const float* wr = wb + ic * 9;
      const float* sp = s_stack + ic * HW;
      #pragma unroll
      for (int ky = 0; ky < 3; ++ky) {
        const int yy = py + ky - 1;
        if (yy < 0 || yy > 15) continue;
        #pragma unroll
        for (int kx = 0; kx < 3; ++kx) {
          const int xx = px + kx - 1;
          if (xx < 0 || xx > 15) continue;
          acc += sp[yy * 16 + xx] * wr[ky * 3 + kx];
        }
      }
    }
    s_ru[oc * HW + tid] = sigmoidf_(acc);
  }
  __syncthreads();   // all gate reads of s_stack done

  #pragma unroll
  for (int c = 0; c < TD_C; ++c) {
    const float v = (c < CIN) ? bu[c] : s_ru[(c - CIN) * HW + tid] * h[c - CIN];
    s_stack[c * HW + tid] = v * mod[c];
  }
  __syncthreads();

  #pragma unroll
  for (int oc = 0; oc < HD; ++oc) {
    float acc = cand_b[oc];
    const float* wb = cand_w + oc * (TD_C * 9);
    for (int ic = 0; ic < TD_C; ++ic) {
      const float* wr = wb + ic * 9;
      const float* sp = s_stack + ic * HW;
      #pragma unroll
      for (int ky = 0; ky < 3; ++ky) {
        const int yy = py + ky - 1;
        if (yy < 0 || yy > 15) continue;
        #pragma unroll
        for (int kx = 0; kx < 3; ++kx) {
          const int xx = px + kx - 1;
          if (xx < 0 || xx > 15) continue;
          acc += sp[yy * 16 + xx] * wr[ky * 3 + kx];
        }
      }
    }
    const float cand = tanhf(acc);
    const float u = s_ru[(HD + oc) * HW + tid];
    h_out[(size_t)b * (HD * HW) + oc * HW + tid] = (1.f - u) * h[oc] + u * cand;
  }
}

// ---------------------------------------------------------------------------
// fc2: out(32,10) = relu'd fc1(32,100) @ fc2_w(10,100)^T + fc2_b
// ---------------------------------------------------------------------------
__global__ void fc2_kernel(const float* __restrict__ a,
                           const float* __restrict__ w,
                           const float* __restrict__ bias,
                           float* __restrict__ out)
{
  const int t = threadIdx.x;
  if (t >= 320) return;
  const int m = t / 10, n = t % 10;
  float acc = bias[n];
  for (int j = 0; j < 100; ++j)
    acc += a[m * 100 + j] * w[n * 100 + j];
  out[m * 10 + n] = acc;
}

// ---------------------------------------------------------------------------
extern "C" void kernel_launch(void* const* d_in, const int* in_sizes, int n_in,
                              void* d_out, int out_size, void* d_ws, size_t ws_size,
                              hipStream_t stream)
{
  const float* x         = (const float*)d_in[0];
  const float* conn      = (const float*)d_in[1];
  const float* conv_in_w = (const float*)d_in[2];
  const float* conv_in_b = (const float*)d_in[3];
  const float* bu_w      = (const float*)d_in[4];
  const float* bu_b      = (const float*)d_in[5];
  const float* td_w      = (const float*)d_in[6];
  const float* td_b      = (const float*)d_in[7];
  const float* gate_w    = (const float*)d_in[8];
  const float* gate_b    = (const float*)d_in[9];
  const float* cand_w    = (const float*)d_in[10];
  const float* cand_b    = (const float*)d_in[11];
  const float* fc1_w     = (const float*)d_in[12];
  const float* fc1_b     = (const float*)d_in[13];
  const float* fc2_w     = (const float*)d_in[14];
  const float* fc2_b     = (const float*)d_in[15];
  float* out = (float*)d_out;
  float* ws  = (float*)d_ws;

  // workspace layout (floats)
  float* cur_in  = ws;                     // 32*4096
  float* st[4]   = { ws + 131072 * 1, ws + 131072 * 2,
                     ws + 131072 * 3, ws + 131072 * 4 };
  float* bu_buf  = ws + 131072 * 5;        // 32*2048
  float* td_buf  = bu_buf + BATCH * N_BU;  // 32*6144
  float* fc1_buf = td_buf + BATCH * N_TD;  // 32*100
  float* part    = fc1_buf + BATCH * 100;  // max(KC_BU*32*2048, KC_TD*32*6144)

  conv_in_kernel<<<BATCH, 256, 0, stream>>>(x, conv_in_w, conv_in_b, cur_in);

  // ---------------- rep 0 (read = prev = zeros, structurally) ----------------
  // node 0: bottomup = cur_in; td input = 0 -> td = td_b[0]
  gemm_wmma_f32<false, false, true, false>
      <<<dim3(N_BU / 64, KC_BU), 256, 0, stream>>>(
      cur_in, bu_w, nullptr, part, N_BU, KDIM, N_BU, nullptr, -1);
  reduce_bias_kernel<<<(BATCH * N_BU + 255) / 256, 256, 0, stream>>>(
      part, bu_b, bu_buf, N_BU, N_BU, KC_BU);
  cell_kernel<<<BATCH, 256, 0, stream>>>(
      bu_buf, nullptr, nullptr, nullptr, td_b,
      gate_w, gate_b, cand_w, cand_b, st[0]);
  // nodes 1..3: bottomup input = 0 -> bu = bu_b[node]; td = td_b[node] (or 0)
  for (int node = 1; node < 4; ++node) {
    cell_kernel<<<BATCH, 256, 0, stream>>>(
        nullptr, bu_b + node * N_BU, nullptr, nullptr,
        (node < 3) ? (td_b + (size_t)node * N_TD) : nullptr,
        gate_w + (size_t)node * 2 * HD * TD_C * 9, gate_b + node * 2 * HD,
        cand_w + (size_t)node * HD * TD_C * 9,     cand_b + node * HD, st[node]);
  }

  // ---------------- rep 1 (read = state, updated sequentially) ---------------
  for (int node = 0; node < 4; ++node) {
    if (node > 0) {
      // bottomup = conn[node-1][node] * st[node-1] (fresh this rep)
      gemm_wmma_f32<false, false, true, true>
          <<<dim3(N_BU / 64, KC_BU), 256, 0, stream>>>(
          st[node - 1], bu_w + (size_t)node * N_BU * KDIM, nullptr, part,
          N_BU, KDIM, N_BU, conn, (node - 1) * 4 + node);
      reduce_bias_kernel<<<(BATCH * N_BU + 255) / 256, 256, 0, stream>>>(
          part, bu_b + node * N_BU, bu_buf, N_BU, N_BU, KC_BU);
    }
    // node 0: bu_buf still holds cur_in @ bu_w[0]^T + bu_b[0] (identical) — reuse.
    const float* tdp = nullptr;
    if (node < 3) {
      // td = conn[node+1][node] * st[node+1] (rep-0 value) @ td_w[node]^T + td_b
      gemm_wmma_f32<false, false, true, true>
          <<<dim3(N_TD / 64, KC_TD), 256, 0, stream>>>(
          st[node + 1], td_w + (size_t)node * N_TD * KDIM, nullptr, part,
          N_TD, KDIM, N_TD, conn, (node + 1) * 4 + node);
      reduce_bias_kernel<<<(BATCH * N_TD + 255) / 256, 256, 0, stream>>>(
          part, td_b + (size_t)node * N_TD, td_buf, N_TD, N_TD, KC_TD);
      tdp = td_buf;
    }
    cell_kernel<<<BATCH, 256, 0, stream>>>(
        bu_buf, nullptr, st[node], tdp, nullptr,
        gate_w + (size_t)node * 2 * HD * TD_C * 9, gate_b + node * 2 * HD,
        cand_w + (size_t)node * HD * TD_C * 9,     cand_b + node * HD, st[node]);
  }

  // ---------------- head: relu(st3) @ fc1^T -> relu -> @ fc2^T ---------------
  gemm_wmma_f32<true, true, false, false><<<dim3(2, 1), 256, 0, stream>>>(
      st[3], fc1_w, fc1_b, fc1_buf, 100, KDIM, 100, nullptr, -1);
  fc2_kernel<<<1, 320, 0, stream>>>(fc1_buf, fc2_w, fc2_b, out);
}